// RoIAlignMax_73349451481353
// MI455X (gfx1250) — compile-verified
//
#include <hip/hip_runtime.h>
#include <math.h>
#include <float.h>
#include <stdint.h>

// Problem constants (match reference setup_inputs)
#define SCALE   0.125f
#define HH      136
#define WW      136
#define CC      512
#define RR      2048
#define AHW     7        // aligned grid 7x7
#define CPB     256      // channels per block (2 blocks per roi)
#define HW      (HH * WW)
#define OUTW    (CPB * 9)   // 2304 f32 outputs per block, contiguous

typedef unsigned int v4u __attribute__((ext_vector_type(4)));
typedef int          v8i __attribute__((ext_vector_type(8)));
typedef int          v4i __attribute__((ext_vector_type(4)));

__global__ __launch_bounds__(CPB) void roialign_maxpool_kernel(
    const float* __restrict__ feat,   // (4, 512, 136, 136)
    const float* __restrict__ rois,   // (2048, 5)
    float* __restrict__ out)          // (2048, 512, 3, 3)
{
    __shared__ int   s_hs[AHW], s_ws[AHW];
    __shared__ float s_hr[AHW], s_wr[AHW], s_vh[AHW], s_vw[AHW];
    __shared__ int   s_b;
    __shared__ __align__(16) float s_out[OUTW];   // staged block output (9 KB)

    const int bx   = blockIdx.x;
    const int r    = bx >> 1;        // roi index
    const int cblk = bx & 1;         // channel half
    const int tid  = threadIdx.x;

    // ROI geometry is separable in h and w: only 7 + 7 entries needed.
    if (tid < AHW) {
        const float* roi = rois + (size_t)r * 5;
        float sx1 = roi[1] * SCALE, sy1 = roi[2] * SCALE;
        float sx2 = roi[3] * SCALE, sy2 = roi[4] * SCALE;
        float bh = fmaxf(sy2 - sy1, 0.0f) * (1.0f / 6.0f);
        float bw = fmaxf(sx2 - sx1, 0.0f) * (1.0f / 6.0f);
        float h = sy1 + (float)tid * bh;
        float w = sx1 + (float)tid * bw;
        int hs = (int)floorf(h); if (hs > HH - 2) hs = HH - 2; if (hs < 0) hs = 0;
        int ws = (int)floorf(w); if (ws > WW - 2) ws = WW - 2; if (ws < 0) ws = 0;
        s_hs[tid] = hs;                s_ws[tid] = ws;
        s_hr[tid] = h - (float)hs;     s_wr[tid] = w - (float)ws;
        s_vh[tid] = (h >= 0.0f && h < (float)HH) ? 1.0f : 0.0f;
        s_vw[tid] = (w >= 0.0f && w < (float)WW) ? 1.0f : 0.0f;
        if (tid == 0) s_b = (int)roi[0];
    }
    __syncthreads();

    // ---- Scalarize the block-uniform part of every gather address. ----
    // Per-lane part: channel offset (elements). Uniform part: batch+row+col.
    const int  b_u   = __builtin_amdgcn_readfirstlane(s_b);
    const float* featb = feat + (size_t)b_u * CC * HW;        // uniform (SGPR)
    const int  cOff  = (cblk * CPB + tid) * HW;               // per-lane, once

    int hsU[AHW], wsU[AHW];
    #pragma unroll
    for (int i = 0; i < AHW; ++i) {
        hsU[i] = __builtin_amdgcn_readfirstlane(s_hs[i]);
        wsU[i] = __builtin_amdgcn_readfirstlane(s_ws[i]);
    }

    // Warm row-pair cachelines (gfx1250 global_prefetch path).
    #pragma unroll
    for (int ph = 0; ph < AHW; ++ph) {
        const float* p0 = featb + hsU[ph] * WW + wsU[0] + cOff;
        __builtin_prefetch(p0, 0, 3);
        __builtin_prefetch(p0 + WW, 0, 3);
    }

    // Bilinear-interpolated 7x7 grid for this (roi, channel), kept in VGPRs.
    float vals[AHW * AHW];
    #pragma unroll
    for (int ph = 0; ph < AHW; ++ph) {
        const float hr  = s_hr[ph];
        const float ihr = 1.0f - hr;
        const float mh  = s_vh[ph];
        const float* rowu = featb + hsU[ph] * WW;   // uniform base (SGPR pair)
        #pragma unroll
        for (int pw = 0; pw < AHW; ++pw) {
            const float* pu = rowu + wsU[pw];       // still uniform
            const float ul  = pu[cOff];             // SADDR + v_chanoff form
            const float ur  = pu[cOff + 1];
            const float dl  = pu[cOff + WW];
            const float dr  = pu[cOff + WW + 1];
            const float wr  = s_wr[pw];
            const float iwr = 1.0f - wr;
            float v = ul * (ihr * iwr) + ur * (ihr * wr)
                    + dl * (hr  * iwr) + dr * (hr  * wr);
            vals[ph * AHW + pw] = v * (mh * s_vw[pw]);
        }
    }

    // 3x3 max-pool, stride 2, VALID -> 9 results, staged in LDS for bulk store.
    float* o = &s_out[tid * 9];
    #pragma unroll
    for (int i = 0; i < 3; ++i) {
        #pragma unroll
        for (int j = 0; j < 3; ++j) {
            float m = -FLT_MAX;
            #pragma unroll
            for (int di = 0; di < 3; ++di) {
                #pragma unroll
                for (int dj = 0; dj < 3; ++dj) {
                    m = fmaxf(m, vals[(2 * i + di) * AHW + (2 * j + dj)]);
                }
            }
            o[i * 3 + j] = m;
        }
    }
    __syncthreads();

    // ---- Bulk output store: one Tensor-Data-Mover DMA per block. ----
    float* gdst = out + ((size_t)r * CC + (size_t)cblk * CPB) * 9;   // 9216 B, 16B-aligned
#if __has_builtin(__builtin_amdgcn_tensor_store_from_lds)
    if (tid == 0) {
        const uint64_t ga = (uint64_t)(uintptr_t)gdst;
        const uint32_t la = (uint32_t)(uintptr_t)&s_out[0];   // generic low 32b = LDS offset
        // D# group0: count=1 | lds_addr | global_addr[56:0] | type=2
        v4u g0 = { 1u,
                   la,
                   (uint32_t)(ga & 0xFFFFFFFFu),
                   (uint32_t)((ga >> 32) & 0x01FFFFFFu) | (2u << 30) };
        // D# group1: data_size=4B; tensor_dim0=tile_dim0=2304; dim1=1 (degenerate)
        v8i g1 = { (int)(2u << 16),                   // w0: mask=0, data_size=2(4B)
                   (int)((OUTW & 0xFFFFu) << 16),     // w1[31:16] = tensor_dim0[15:0]
                   (int)(1u << 16),                   // w2: dim0[31:16]=0, tensor_dim1=1
                   (int)((OUTW & 0xFFFFu) << 16),     // w3[31:16] = tile_dim0
                   1,                                 // w4: tile_dim1=1, tile_dim2=0
                   OUTW,                              // w5: tensor_dim0_stride[31:0]
                   0, 0 };                            // w6,w7: stride hi / dim1 stride
        v4i g2 = { 0, 0, 0, 0 };   // normal mode, dims 2/3 unused
        v4i g3 = { 0, 0, 0, 0 };   // normal mode, dim 4 unused
        v8i g4 = { 0, 0, 0, 0, 0, 0, 0, 0 };   // extra group (clang-23 6-arg form)
        __builtin_amdgcn_tensor_store_from_lds(g0, g1, g2, g3, g4, 0);
#if __has_builtin(__builtin_amdgcn_s_wait_tensorcnt)
        __builtin_amdgcn_s_wait_tensorcnt(0);
#endif
    }
#else
    // Fallback: coalesced float4 copy LDS -> global (576 x b128 per block).
    const float4* src4 = (const float4*)&s_out[0];
    float4* dst4 = (float4*)gdst;
    #pragma unroll
    for (int i = tid; i < OUTW / 4; i += CPB) {
        dst4[i] = src4[i];
    }
#endif
}

extern "C" void kernel_launch(void* const* d_in, const int* in_sizes, int n_in,
                              void* d_out, int out_size, void* d_ws, size_t ws_size,
                              hipStream_t stream) {
    const float* feat = (const float*)d_in[0];   // (4,512,136,136) f32
    const float* rois = (const float*)d_in[1];   // (2048,5) f32
    float* out = (float*)d_out;                  // (2048,512,3,3) f32

    dim3 grid(RR * (CC / CPB));   // 4096 blocks
    dim3 block(CPB);              // 256 threads = 8 wave32
    roialign_maxpool_kernel<<<grid, block, 0, stream>>>(feat, rois, out);
}